// Blur3d_40973988004557
// MI455X (gfx1250) — compile-verified
//
#include <hip/hip_runtime.h>

// ---- CDNA5 (gfx1250) blur3d:
//  Phase 1: global->LDS via global_load_async_to_lds_b128 (ASYNCcnt)
//  Phase 2: VALU sliding-window horizontal pass (separable taps)
//  Phase 3: vertical pass as WMMA banded matmul (v_wmma_f32_16x16x4_f32)
//  Phase 4: LDS->global via global_store_async_from_lds_b128 (512B rows)

typedef float v2f __attribute__((ext_vector_type(2)));
typedef float v8f __attribute__((ext_vector_type(8)));
typedef float f4v __attribute__((ext_vector_type(4)));
typedef int   v4i __attribute__((ext_vector_type(4)));

#define AS1 __attribute__((address_space(1)))
#define AS3 __attribute__((address_space(3)))

#ifndef __has_builtin
#define __has_builtin(x) 0
#endif

#if __has_builtin(__builtin_amdgcn_global_load_async_to_lds_b128)
#define HAVE_ASYNC_LD 1
#else
#define HAVE_ASYNC_LD 0
#endif

#if __has_builtin(__builtin_amdgcn_global_store_async_from_lds_b128)
#define HAVE_ASYNC_ST 1
#else
#define HAVE_ASYNC_ST 0
#endif

#define IMG_W 128
#define IMG_H 128
#define STRIDE 132                       // row pitch in floats: 528 B, 16B-aligned
#define SMEM_FLOATS (2 * IMG_H * STRIDE) // raw image + H intermediate (~132 KB)

__global__ __launch_bounds__(256) void blur3d_wmma_kernel(
    const float* __restrict__ in, const float* __restrict__ kern,
    float* __restrict__ out)
{
    extern __shared__ float smem[];
    float* in_s = smem;                  // [128][STRIDE] raw image / out staging
    float* h_s  = smem + IMG_H * STRIDE; // [128][STRIDE] horizontally filtered

    const int t = threadIdx.x;
    const size_t base = (size_t)blockIdx.x * (size_t)(IMG_W * IMG_H);
    const float* gsrc = in + base;

    // ---- separable taps from the (rank-1, sum==1) 4x4 kernel ----
    // k2d[a][b] == r[a]*c[b] where r = row sums, c = col sums.
    float k2[16];
#pragma unroll
    for (int i = 0; i < 16; ++i) k2[i] = kern[i];
    const float r0t = k2[0]  + k2[1]  + k2[2]  + k2[3];
    const float r1t = k2[4]  + k2[5]  + k2[6]  + k2[7];
    const float r2t = k2[8]  + k2[9]  + k2[10] + k2[11];
    const float r3t = k2[12] + k2[13] + k2[14] + k2[15];
    const float c0t = k2[0]  + k2[4]  + k2[8]  + k2[12];
    const float c1t = k2[1]  + k2[5]  + k2[9]  + k2[13];
    const float c2t = k2[2]  + k2[6]  + k2[10] + k2[14];
    const float c3t = k2[3]  + k2[7]  + k2[11] + k2[15];

    // ---- Phase 1: 64 KB image -> LDS, 16B per lane per issue ----
#pragma unroll
    for (int i = 0; i < 16; ++i) {
        const int c  = i * 256 + t;      // 16-byte chunk id, 4096 total
        const int yy = c >> 5;           // image row
        const int xx = (c & 31) << 2;    // float offset within row
#if HAVE_ASYNC_LD
        __builtin_amdgcn_global_load_async_to_lds_b128(
            (AS1 v4i*)(gsrc + (c << 2)),
            (AS3 v4i*)(&in_s[yy * STRIDE + xx]), 0, 0);
#else
        f4v v = *(const f4v*)(gsrc + (c << 2));
        *(f4v*)(&in_s[yy * STRIDE + xx]) = v;
#endif
    }
#if HAVE_ASYNC_LD
#if __has_builtin(__builtin_amdgcn_s_wait_asynccnt)
    __builtin_amdgcn_s_wait_asynccnt(0);
#else
    asm volatile("s_wait_asynccnt 0" ::: "memory");
#endif
#endif
    __syncthreads();

    // ---- Phase 2: horizontal pass (true conv): ----
    // H[y][x] = c3*in[x-2] + c2*in[x-1] + c1*in[x] + c0*in[x+1], zero-padded.
    {
        const int y  = t & 127;
        const int x0 = (t >> 7) * 64;    // two threads per row, 64 cols each
        const float* row = &in_s[y * STRIDE];
        float w0 = (x0 >= 2) ? row[x0 - 2] : 0.f;
        float w1 = (x0 >= 1) ? row[x0 - 1] : 0.f;
        float w2 = row[x0];
        float w3 = row[x0 + 1];
#pragma unroll 4
        for (int x = x0; x < x0 + 64; ++x) {
            h_s[y * STRIDE + x] = c3t * w0 + c2t * w1 + c1t * w2 + c0t * w3;
            float nxt = (x + 2 < IMG_W) ? row[x + 2] : 0.f;
            w0 = w1; w1 = w2; w2 = w3; w3 = nxt;
        }
    }
    __syncthreads();        // after this point in_s is dead -> reuse for output

    // ---- Phase 3: vertical pass as WMMA banded matmul ----
    // Out[m][n] = sum_a r[a] * H[y0+m+1-a][n]  (a in 0..3)
    // In-block part: Out_tile = A_band x H_tile, A_band[m][u] = r[m+1-u].
    const int lane = t & 31;
    const int wv   = t >> 5;             // wave id 0..7 -> tile row
    const int m    = lane & 15;          // A: row M / B: col N / fixups: col n
    const int koff = (lane >> 4) << 1;   // 0 (lanes 0-15) or 2 (lanes 16-31)

    // A fragments (per-lane, built once): chunk kb holds K cols 4kb..4kb+3.
    v2f afr[4];
#pragma unroll
    for (int kb = 0; kb < 4; ++kb) {
#pragma unroll
        for (int tt = 0; tt < 2; ++tt) {
            const int col = 4 * kb + koff + tt;
            const int d   = m + 1 - col;
            afr[kb][tt] = (d == 0) ? r0t : (d == 1) ? r1t
                        : (d == 2) ? r2t : (d == 3) ? r3t : 0.f;
        }
    }

    const int y0  = wv * 16;
    const int ym2 = (y0 >= 2) ? (y0 - 2) : 0;           // clamped halo rows
    const int ym1 = (y0 >= 1) ? (y0 - 1) : 0;
    const int yp1 = (y0 + 16 < IMG_H) ? (y0 + 16) : 0;
    // Halo fix-up coefficients (zeroed on wrong half-wave / image edge):
    const bool lo = (lane < 16);
    const float g0 = (lo && y0 >= 2)            ? r3t : 0.f; // m=0  <- H[y0-2]
    const float g1 = (lo && y0 >= 1)            ? r2t : 0.f; // m=0  <- H[y0-1]
    const float g2 = (lo && y0 >= 1)            ? r3t : 0.f; // m=1  <- H[y0-1]
    const float g3 = (!lo && (y0 + 16 < IMG_H)) ? r0t : 0.f; // m=15 <- H[y0+16]

#pragma unroll
    for (int tx = 0; tx < 8; ++tx) {
        const int x0t = tx * 16;
        v8f acc = {};
#pragma unroll
        for (int kb = 0; kb < 4; ++kb) {
            const int rrow = y0 + 4 * kb + koff; // K rows for this lane half
            v2f bfr;
            bfr[0] = h_s[rrow * STRIDE + x0t + m];
            bfr[1] = h_s[(rrow + 1) * STRIDE + x0t + m];
            acc = __builtin_amdgcn_wmma_f32_16x16x4_f32(
                false, afr[kb], false, bfr, (short)0, acc, false, false);
        }
        // Rank-3 halo fix-ups (predicated VALU, EXEC stays all-ones):
        const float hm2 = h_s[ym2 * STRIDE + x0t + m];
        const float hm1 = h_s[ym1 * STRIDE + x0t + m];
        const float hp1 = h_s[yp1 * STRIDE + x0t + m];
        acc[0] += g0 * hm2 + g1 * hm1;
        acc[1] += g2 * hm1;
        acc[7] += g3 * hp1;

        // C/D layout -> vgpr i: lanes 0-15: M=i ; lanes 16-31: M=8+i.
        const int rbase = y0 + (lo ? 0 : 8);
#if HAVE_ASYNC_ST
        // Stage into in_s (dead buffer) for wide async drain.
#pragma unroll
        for (int i = 0; i < 8; ++i)
            in_s[(rbase + i) * STRIDE + x0t + m] = acc[i];
#else
#pragma unroll
        for (int i = 0; i < 8; ++i)
            out[base + (size_t)(rbase + i) * IMG_W + x0t + m] = acc[i];
#endif
    }

#if HAVE_ASYNC_ST
    // ---- Phase 4: drain staged image LDS -> global, 512B-contiguous rows ----
    __syncthreads();                     // staging rows come from other waves
    float* gdst = out + base;
#pragma unroll
    for (int i = 0; i < 16; ++i) {
        const int c  = i * 256 + t;      // 16-byte chunk id, 4096 total
        const int yy = c >> 5;
        const int xx = (c & 31) << 2;
        __builtin_amdgcn_global_store_async_from_lds_b128(
            (AS1 v4i*)(gdst + (c << 2)),
            (AS3 v4i*)(&in_s[yy * STRIDE + xx]), 0, 0);
    }
    // S_ENDPGM performs an implicit wait-idle (ASYNCcnt drains before exit).
#endif
}

extern "C" void kernel_launch(void* const* d_in, const int* in_sizes, int n_in,
                              void* d_out, int out_size, void* d_ws, size_t ws_size,
                              hipStream_t stream) {
    const float* x = (const float*)d_in[0];
    const float* k = (const float*)d_in[1];
    float* o = (float*)d_out;
    const int n_img = in_sizes[0] / (IMG_W * IMG_H); // 4096 images
    const size_t smem = (size_t)SMEM_FLOATS * sizeof(float); // ~132 KB (< 320 KB/WGP)
    blur3d_wmma_kernel<<<n_img, 256, smem, stream>>>(x, k, o);
}